// YOWO_61856118997001
// MI455X (gfx1250) — compile-verified
//
#include <hip/hip_runtime.h>
#include <math.h>

typedef __attribute__((ext_vector_type(2))) float v2f;
typedef __attribute__((ext_vector_type(8))) float v8f;
typedef __attribute__((ext_vector_type(4))) unsigned int v4u;
typedef __attribute__((ext_vector_type(4))) int v4i;
typedef __attribute__((ext_vector_type(8))) int v8i;

#define HEAD_DIM     256
#define NUM_CLASSES  80
#define BATCH        32
#define TOPK_K       1000
#define CAND_MAX     2048
#define TOTAL_ANCH   5376
#define OUT_BOX_F    (BATCH * 3 * TOPK_K * 6)    /* 576000 floats */
#define POOL_F       (3 * BATCH * 256 * 16 * 16) /* 6291456 floats */

#if __has_builtin(__builtin_amdgcn_tensor_load_to_lds) && __has_builtin(__builtin_amdgcn_s_wait_tensorcnt)
#define HAVE_TDM 1
#endif
#if __has_builtin(__builtin_amdgcn_global_load_async_to_lds_b128) && __has_builtin(__builtin_amdgcn_s_wait_asynccnt)
#define HAVE_ASYNC 1
#endif

__device__ __forceinline__ float sigmoidf_(float x) { return 1.0f / (1.0f + expf(-x)); }

/* ---- init: default output rows (0,0,0,0,0,-1) + zero candidate counters ---- */
__global__ void k_init(float* __restrict__ out, int* __restrict__ candCount) {
    int i = blockIdx.x * blockDim.x + threadIdx.x;
    if (i < OUT_BOX_F) out[i] = ((i % 6) == 5) ? -1.0f : 0.0f;
    if (i < 96) candCount[i] = 0;
}

/* ---- pack weights into exact WMMA A-operand lane order + padded bias ----
 * Wt layout: [lvl][mtile(6)][kstep(64)][lane(32)][2]  (row = mtile*16 + lane&15,
 * k = kstep*4 + (lane>>4)*2 + j)  -> inner-loop load is one coalesced v2f/lane. */
__global__ void k_pack(const float* __restrict__ conf_w, const float* __restrict__ conf_b,
                       const float* __restrict__ cls_w,  const float* __restrict__ cls_b,
                       const float* __restrict__ reg_w,  const float* __restrict__ reg_b,
                       float* __restrict__ Wt, float* __restrict__ biasPad) {
    int i = blockIdx.x * blockDim.x + threadIdx.x;
    if (i < 3 * 96) {
        int lvl = i / 96, row = i % 96;
        float bv = 0.0f;
        if (row < 80)       bv = cls_b[lvl * 80 + row];
        else if (row == 80) bv = conf_b[lvl];
        else if (row < 85)  bv = reg_b[lvl * 4 + (row - 81)];
        biasPad[i] = bv;
    }
    if (i < 3 * 6 * 64 * 64) {
        int lane2 = i & 63;
        int kstep = (i >> 6) & 63;
        int m     = (i >> 12) % 6;
        int lvl   = i / (4096 * 6);
        int lane  = lane2 >> 1, jj = lane2 & 1;
        int row   = m * 16 + (lane & 15);
        int k     = kstep * 4 + ((lane >> 4) << 1) + jj;
        float v = 0.0f;
        if (row < 80)       v = cls_w[(lvl * 80 + row) * HEAD_DIM + k];
        else if (row == 80) v = conf_w[lvl * HEAD_DIM + k];
        else if (row < 85)  v = reg_w[(lvl * 4 + (row - 81)) * HEAD_DIM + k];
        Wt[i] = v;
    }
}

/* ---- fused head: TDM-staged feature tile + WMMA fp32 GEMM (96x256 @ 256x16)
 * + sigmoid-score thresholding (candidate compaction) + box decode ---- */
__global__ __launch_bounds__(192) void k_head(
    const float* __restrict__ feat, const float* __restrict__ Wt,
    const float* __restrict__ biasPad, float* __restrict__ boxes,
    float* __restrict__ cand, int* __restrict__ candCount,
    int lvl, int W, int HW, float stride, int anchorOffset)
{
    __shared__ __align__(16) float Bsh[256 * 16];  /* feature tile K x N */
    __shared__ __align__(16) float Dsh[96 * 16];   /* output tile  M x N */
    const int tid  = threadIdx.x;
    const int wave = tid >> 5;
    const int lane = tid & 31;
    const int b    = blockIdx.y;
    const int tileN = blockIdx.x * 16;

    const float* fb = feat + (size_t)b * HEAD_DIM * HW;

#if defined(HAVE_TDM)
    /* One TDM descriptor loads the whole 16x256 fp32 tile (16KB) into LDS:
     * 2D tile: tile_dim0=16 (contiguous cols), tile_dim1=256 (rows, stride HW) */
    if (wave == 0) {
        unsigned long long ga = (unsigned long long)(uintptr_t)(fb + tileN);
        unsigned int la = (unsigned int)(uintptr_t)(&Bsh[0]); /* low 32 bits = LDS offset */
        v4u g0;
        g0[0] = 1u;                                   /* count=1, user descriptor */
        g0[1] = la;                                   /* lds_addr */
        g0[2] = (unsigned int)(ga & 0xFFFFFFFFu);     /* global_addr[31:0] */
        g0[3] = (unsigned int)((ga >> 32) & 0x01FFFFFFu) | (2u << 30); /* addr[56:32] | type=2 */
        v8i g1;
        g1[0] = (int)(2u << 16);                      /* data_size=4B */
        g1[1] = (int)(((unsigned int)HW & 0xFFFFu) << 16);        /* tensor_dim0[15:0] */
        g1[2] = (int)((((unsigned int)HW >> 16) & 0xFFFFu) | (256u << 16)); /* dim0[31:16] | dim1[15:0] */
        g1[3] = (int)(16u << 16);                     /* dim1[31:16]=0 | tile_dim0=16 */
        g1[4] = 256;                                  /* tile_dim1=256 | tile_dim2=0 */
        g1[5] = HW;                                   /* tensor_dim0_stride[31:0] */
        g1[6] = 0;
        g1[7] = 0;
        v4i gz; gz[0] = 0; gz[1] = 0; gz[2] = 0; gz[3] = 0;
        v8i gz8; gz8[0] = 0; gz8[1] = 0; gz8[2] = 0; gz8[3] = 0;
        gz8[4] = 0; gz8[5] = 0; gz8[6] = 0; gz8[7] = 0;
        __builtin_amdgcn_tensor_load_to_lds(g0, g1, gz, gz, gz8, 0);
        __builtin_amdgcn_s_wait_tensorcnt(0);
    }
    __syncthreads();
#elif defined(HAVE_ASYNC)
    {
        const float* src = fb + tileN;
        for (int ci = tid; ci < 1024; ci += 192) {       /* 1024 x 16B chunks */
            int k  = ci >> 2;
            int n4 = (ci & 3) << 2;
            unsigned long long gaddr = (unsigned long long)(uintptr_t)(src + (size_t)k * HW + n4);
            unsigned int laddr = (unsigned int)(uintptr_t)(&Bsh[ci << 2]);
            __builtin_amdgcn_global_load_async_to_lds_b128(
                (__attribute__((address_space(1))) void*)(uintptr_t)gaddr,
                (__attribute__((address_space(3))) void*)laddr, 0, 0);
        }
        __builtin_amdgcn_s_wait_asynccnt(0);
    }
    __syncthreads();
#else
    for (int i = tid; i < 256 * 16; i += 192) {
        int k = i >> 4, n = i & 15;
        Bsh[i] = fb[(size_t)k * HW + tileN + n];
    }
    __syncthreads();
#endif

    const int lane16 = lane & 15;
    const int half   = lane >> 4;
    const float* wbase = Wt + (size_t)(lvl * 6 + wave) * 64 * 64;

    v8f c = {};
#pragma unroll 8
    for (int kstep = 0; kstep < 64; ++kstep) {
        v2f a = *(const v2f*)(wbase + kstep * 64 + lane * 2);
        int k0 = kstep * 4 + (half << 1);
        v2f bb;
        bb.x = Bsh[k0 * 16 + lane16];
        bb.y = Bsh[(k0 + 1) * 16 + lane16];
        c = __builtin_amdgcn_wmma_f32_16x16x4_f32(false, a, false, bb,
                                                  (short)0, c, false, false);
    }
#pragma unroll
    for (int i = 0; i < 8; ++i)
        Dsh[(wave * 16 + i + half * 8) * 16 + lane16] = c[i];
    __syncthreads();

    const float* bias = biasPad + lvl * 96;

    /* fused score + rare-candidate compaction (score > 0.1) */
    for (int i = tid; i < 16 * NUM_CLASSES; i += 192) {
        int n = i / NUM_CLASSES, cls = i % NUM_CLASSES;
        float confv = Dsh[80 * 16 + n] + bias[80];
        float clsv  = Dsh[cls * 16 + n] + bias[cls];
        float s = sqrtf(sigmoidf_(confv) * sigmoidf_(clsv));
        if (s > 0.1f) {
            int list = b * 3 + lvl;
            int slot = atomicAdd(&candCount[list], 1);
            if (slot < CAND_MAX) {
                int base = (list * CAND_MAX + slot) * 2;
                cand[base] = s;
                ((int*)cand)[base + 1] = (tileN + n) * NUM_CLASSES + cls;
            }
        }
    }

    /* box decode: rows 81..84 are reg */
    if (tid < 16) {
        int n = tid, anchor = tileN + n;
        float rx = Dsh[81 * 16 + n] + bias[81];
        float ry = Dsh[82 * 16 + n] + bias[82];
        float rw = Dsh[83 * 16 + n] + bias[83];
        float rh = Dsh[84 * 16 + n] + bias[84];
        float ax = ((float)(anchor % W) + 0.5f) * stride;
        float ay = ((float)(anchor / W) + 0.5f) * stride;
        float cx = ax + rx * stride, cy = ay + ry * stride;
        float hw = 0.5f * expf(rw) * stride, hh = 0.5f * expf(rh) * stride;
        float* bo = boxes + ((size_t)b * TOTAL_ANCH + anchorOffset + anchor) * 4;
        bo[0] = cx - hw; bo[1] = cy - hh; bo[2] = cx + hw; bo[3] = cy + hh;
    }
}

/* ---- per-(b,lvl) bitonic sort of candidates, emit sorted top-k rows ---- */
__global__ __launch_bounds__(1024) void k_topk(
    const float* __restrict__ cand, const int* __restrict__ candCount,
    const float* __restrict__ boxes, float* __restrict__ out)
{
    __shared__ float skey[CAND_MAX];
    __shared__ int   sidx[CAND_MAX];
    const int list = blockIdx.x;          /* b*3 + lvl */
    const int b = list / 3, lvl = list % 3;
    int count = candCount[list];
    if (count > CAND_MAX) count = CAND_MAX;
    const int tid = threadIdx.x;

    for (int i = tid; i < CAND_MAX; i += 1024) {
        if (i < count) {
            skey[i] = cand[(list * CAND_MAX + i) * 2];
            sidx[i] = ((const int*)cand)[(list * CAND_MAX + i) * 2 + 1];
        } else {
            skey[i] = -1.0f;
            sidx[i] = -1;
        }
    }
    __syncthreads();

    for (int k = 2; k <= CAND_MAX; k <<= 1) {
        for (int j = k >> 1; j > 0; j >>= 1) {
            for (int i = tid; i < CAND_MAX; i += 1024) {
                int ixj = i ^ j;
                if (ixj > i) {
                    bool descRegion = ((i & k) == 0);
                    float a = skey[i], bk = skey[ixj];
                    bool doSwap = descRegion ? (a < bk) : (a > bk);
                    if (doSwap) {
                        skey[i] = bk; skey[ixj] = a;
                        int t = sidx[i]; sidx[i] = sidx[ixj]; sidx[ixj] = t;
                    }
                }
            }
            __syncthreads();
        }
    }

    const int anchorOffset = (lvl == 0) ? 0 : ((lvl == 1) ? 4096 : 5120);
    if (tid < TOPK_K && tid < count) {
        float s = skey[tid];
        if (s > 0.1f) {
            int idx = sidx[tid];
            int anchor = idx / NUM_CLASSES;
            int cls = idx - anchor * NUM_CLASSES;
            const float* bo = boxes + ((size_t)b * TOTAL_ANCH + anchorOffset + anchor) * 4;
            float* o = out + ((size_t)(b * 3 + lvl) * TOPK_K + tid) * 6;
#pragma unroll
            for (int q = 0; q < 4; ++q) {
                float v = bo[q] * (1.0f / 512.0f);
                o[q] = fminf(fmaxf(v, 0.0f), 1.0f);
            }
            o[4] = s;
            o[5] = (float)cls;
        }
    }
}

/* ---- pooled output: avgpool 4/2/1 -> [3,B,256,16,16] ---- */
__global__ void k_pool(const float* __restrict__ f0, const float* __restrict__ f1,
                       const float* __restrict__ f2, float* __restrict__ out)
{
    int idx = blockIdx.x * blockDim.x + threadIdx.x;
    if (idx >= POOL_F) return;
    int x = idx & 15, y = (idx >> 4) & 15;
    int c = (idx >> 8) & 255, b = (idx >> 16) & 31, l = idx >> 21;
    float v;
    if (l == 0) {
        const float* p = f0 + (((size_t)b * 256 + c) * 64 + y * 4) * 64 + x * 4;
        float s = 0.0f;
#pragma unroll
        for (int dy = 0; dy < 4; ++dy)
#pragma unroll
            for (int dx = 0; dx < 4; ++dx) s += p[dy * 64 + dx];
        v = s * (1.0f / 16.0f);
    } else if (l == 1) {
        const float* p = f1 + (((size_t)b * 256 + c) * 32 + y * 2) * 32 + x * 2;
        v = (p[0] + p[1] + p[32] + p[33]) * 0.25f;
    } else {
        v = f2[(((size_t)b * 256 + c) * 16 + y) * 16 + x];
    }
    out[OUT_BOX_F + idx] = v;
}

extern "C" void kernel_launch(void* const* d_in, const int* in_sizes, int n_in,
                              void* d_out, int out_size, void* d_ws, size_t ws_size,
                              hipStream_t stream) {
    const float* feat0  = (const float*)d_in[0];
    const float* feat1  = (const float*)d_in[1];
    const float* feat2  = (const float*)d_in[2];
    const float* conf_w = (const float*)d_in[3];
    const float* conf_b = (const float*)d_in[4];
    const float* cls_w  = (const float*)d_in[5];
    const float* cls_b  = (const float*)d_in[6];
    const float* reg_w  = (const float*)d_in[7];
    const float* reg_b  = (const float*)d_in[8];
    float* out = (float*)d_out;

    float* ws       = (float*)d_ws;
    float* Wt       = ws;                 /* 3*6*64*64   = 73728  */
    float* biasPad  = ws + 73728;         /* 3*96        = 288    */
    float* boxes    = ws + 74016;         /* 32*5376*4   = 688128 */
    int*   candCnt  = (int*)(ws + 762144);/* 96 ints              */
    float* cand     = ws + 762240;        /* 96*2048*2   = 393216 */

    k_init<<<(OUT_BOX_F + 255) / 256, 256, 0, stream>>>(out, candCnt);
    k_pack<<<(73728 + 255) / 256, 256, 0, stream>>>(conf_w, conf_b, cls_w, cls_b,
                                                    reg_w, reg_b, Wt, biasPad);

    dim3 g0(4096 / 16, BATCH), g1(1024 / 16, BATCH), g2(256 / 16, BATCH);
    k_head<<<g0, 192, 0, stream>>>(feat0, Wt, biasPad, boxes, cand, candCnt,
                                   0, 64, 4096, 8.0f, 0);
    k_head<<<g1, 192, 0, stream>>>(feat1, Wt, biasPad, boxes, cand, candCnt,
                                   1, 32, 1024, 16.0f, 4096);
    k_head<<<g2, 192, 0, stream>>>(feat2, Wt, biasPad, boxes, cand, candCnt,
                                   2, 16, 256, 32.0f, 5120);

    k_topk<<<96, 1024, 0, stream>>>(cand, candCnt, boxes, out);
    k_pool<<<(POOL_F + 255) / 256, 256, 0, stream>>>(feat0, feat1, feat2, out);
}